// PAConv_32263794328108
// MI455X (gfx1250) — compile-verified
//
#include <hip/hip_runtime.h>
#include <math.h>

// ---------------- problem constants ----------------
#define Bsz  4
#define Npts 2048
#define Kn   30
#define ROWS (Bsz*Npts)          // 8192

typedef float v2f __attribute__((ext_vector_type(2)));
typedef float v8f __attribute__((ext_vector_type(8)));

// ---------------- workspace layout (4-byte elements) ----------------
#define OFF_PTS     0           // 24576   (B,N,3)
#define OFF_IDX     24576       // 245760  (B,N,K) int
#define OFF_XYZ10   270336      // 2457600 (B,N,K,10)
#define OFF_SC      2727936     // 1966080 (B,N,K,8)
#define OFF_SCSUM   4694016     // 65536   (B,N,8)
#define OFF_PO      4759552     // 4194304 (B,N,512)
#define OFF_CO      8953856     // 4194304 (B,N,512)
#define OFF_PRE     13180928    // 524288  (B,N,64)
#define OFF_XX      13705216    // 2621440 (B,N,320)
#define OFF_HT      16326656    // 8388608 (B,N,1024)
#define OFF_CLS     24715264    // 4352    (B,1088)
#define OFF_H6      24719616    // 2097152
#define OFF_H7      26816768    // 2097152
#define OFF_H8      28913920    // 1048576
#define OFF_STATS   29962496    // 3616
#define OFF_SS      29966112    // 512   (bn scale/shift for fused GEMM A-loads)
#define OFF_WT      29966624    // 32768 (Wsum transposed to (out=512,in=64))
#define OFF_CT      29999392    // 32768 (kern_top transposed to (out=512,in=64))
// stats sub-offsets (sum at +0, sumsq at +C)
#define ST_BN1 0
#define ST_SN  128
#define ST_PA  160
#define ST_T   288
#define ST_6   2336
#define ST_7   2848
#define ST_8   3360

#define NEG_INF (-3.4e38f)

// ---------------- small prep kernels ----------------
extern "C" __global__ void k_transpose(const float* __restrict__ x, float* __restrict__ pts, int total) {
  int g = blockIdx.x * blockDim.x + threadIdx.x;
  if (g >= total) return;
  int b = g / (3 * Npts);
  int rem = g % (3 * Npts);
  int c = rem / Npts;
  int n = rem % Npts;
  pts[(b * Npts + n) * 3 + c] = x[g];
}

// one block per (b,n) row: full 2048-way neg-distance + iterative top-30 (ties -> lowest index)
extern "C" __global__ void k_knn(const float* __restrict__ pts, int* __restrict__ idx) {
  __shared__ float dist[Npts];
  __shared__ float bv[256];
  __shared__ int   bi[256];
  int row = blockIdx.x;                 // b*N + n
  int b = row / Npts;
  int n = row % Npts;
  int t = threadIdx.x;
  const float* pb = pts + (size_t)b * Npts * 3;
  float pnx = pb[n*3+0], pny = pb[n*3+1], pnz = pb[n*3+2];
  float sqn = pnx*pnx + pny*pny + pnz*pnz;
  for (int u = 0; u < Npts/256; ++u) {
    int m = t + u*256;
    float mx = pb[m*3+0], my = pb[m*3+1], mz = pb[m*3+2];
    float sqm = mx*mx + my*my + mz*mz;
    dist[m] = 2.f*(pnx*mx + pny*my + pnz*mz) - sqn - sqm;
  }
  __syncthreads();
  for (int kk = 0; kk < Kn; ++kk) {
    float best = NEG_INF; int bidx = 0x7fffffff;
    for (int u = 0; u < Npts/256; ++u) {
      int m = t + u*256;
      float v = dist[m];
      if (v > best || (v == best && m < bidx)) { best = v; bidx = m; }
    }
    bv[t] = best; bi[t] = bidx;
    __syncthreads();
    for (int s = 128; s > 0; s >>= 1) {
      if (t < s) {
        if (bv[t+s] > bv[t] || (bv[t+s] == bv[t] && bi[t+s] < bi[t])) { bv[t] = bv[t+s]; bi[t] = bi[t+s]; }
      }
      __syncthreads();
    }
    if (t == 0) { idx[row*Kn + kk] = bi[0]; dist[bi[0]] = NEG_INF; }
    __syncthreads();
  }
}

extern "C" __global__ void k_xyz10(const float* __restrict__ pts, const int* __restrict__ idx,
                                   float* __restrict__ xyz10, int total) {
  int r = blockIdx.x * blockDim.x + threadIdx.x;
  if (r >= total) return;
  int b = r / (Npts * Kn);
  int n = (r / Kn) % Npts;
  int j = idx[r];
  const float* pc = pts + ((size_t)b * Npts + n) * 3;
  const float* pn = pts + ((size_t)b * Npts + j) * 3;
  float rx = pn[0]-pc[0], ry = pn[1]-pc[1], rz = pn[2]-pc[2];
  float d = sqrtf(rx*rx + ry*ry + rz*rz + 1e-12f);
  float* o = xyz10 + (size_t)r * 10;
  o[0]=pc[0]; o[1]=pc[1]; o[2]=pc[2];
  o[3]=pn[0]; o[4]=pn[1]; o[5]=pn[2];
  o[6]=rx;    o[7]=ry;    o[8]=rz;    o[9]=d;
}

// ---------------- conv1 (edge conv, 6 in-channels, 64 out) ----------------
extern "C" __global__ void k_conv1_stats(const float* __restrict__ xyz10, const float* __restrict__ W,
                                         const float* __restrict__ bias, float* __restrict__ stats) {
  __shared__ float ssum[256], ssq[256];
  int o = blockIdx.x, t = threadIdx.x;
  float w0=W[o*6+0], w1=W[o*6+1], w2=W[o*6+2], w3=W[o*6+3], w4=W[o*6+4], w5=W[o*6+5];
  float bb = bias[o];
  float sum = 0.f, sq = 0.f;
  for (int r = t; r < ROWS*Kn; r += 256) {
    const float* xr = xyz10 + (size_t)r * 10;
    float h = bb + xr[6]*w0 + xr[7]*w1 + xr[8]*w2 + xr[0]*w3 + xr[1]*w4 + xr[2]*w5;
    sum += h; sq += h*h;
  }
  ssum[t] = sum; ssq[t] = sq;
  __syncthreads();
  for (int s = 128; s > 0; s >>= 1) {
    if (t < s) { ssum[t] += ssum[t+s]; ssq[t] += ssq[t+s]; }
    __syncthreads();
  }
  if (t == 0) { stats[o] = ssum[0]; stats[64 + o] = ssq[0]; }
}

extern "C" __global__ void k_conv1_apply(const float* __restrict__ xyz10, const float* __restrict__ W,
                                         const float* __restrict__ bias, const float* __restrict__ stats,
                                         const float* __restrict__ g, const float* __restrict__ be,
                                         float* __restrict__ xx) {
  int row = blockIdx.x * blockDim.x + threadIdx.x;
  if (row >= ROWS) return;
  const float inv = 1.f / (float)(ROWS * Kn);
  for (int o = 0; o < 64; ++o) {
    float mu = stats[o] * inv;
    float var = stats[64+o] * inv - mu*mu;
    float sc = g[o] * rsqrtf(var + 1e-5f);
    float sh = be[o] - mu * sc;
    float w0=W[o*6+0], w1=W[o*6+1], w2=W[o*6+2], w3=W[o*6+3], w4=W[o*6+4], w5=W[o*6+5];
    float bb = bias[o];
    float mx = NEG_INF;
    for (int k = 0; k < Kn; ++k) {
      const float* xr = xyz10 + ((size_t)row * Kn + k) * 10;
      float h = bb + xr[6]*w0 + xr[7]*w1 + xr[8]*w2 + xr[0]*w3 + xr[1]*w4 + xr[2]*w5;
      float y = h * sc + sh;
      y = y > 0.f ? y : 0.f;
      mx = y > mx ? y : mx;
    }
    xx[(size_t)row * 320 + o] = mx;
  }
}

// ---------------- scorenet ----------------
extern "C" __global__ void k_sn_stats(const float* __restrict__ xyz10, const float* __restrict__ w1,
                                      float* __restrict__ stats) {
  __shared__ float ssum[256], ssq[256];
  int o = blockIdx.x, t = threadIdx.x;
  float w[10];
  for (int c = 0; c < 10; ++c) w[c] = w1[o*10 + c];
  float sum = 0.f, sq = 0.f;
  for (int r = t; r < ROWS*Kn; r += 256) {
    const float* xr = xyz10 + (size_t)r * 10;
    float h = 0.f;
    for (int c = 0; c < 10; ++c) h += xr[c] * w[c];
    sum += h; sq += h*h;
  }
  ssum[t] = sum; ssq[t] = sq;
  __syncthreads();
  for (int s = 128; s > 0; s >>= 1) {
    if (t < s) { ssum[t] += ssum[t+s]; ssq[t] += ssq[t+s]; }
    __syncthreads();
  }
  if (t == 0) { stats[o] = ssum[0]; stats[16 + o] = ssq[0]; }
}

extern "C" __global__ void k_sn_apply(const float* __restrict__ xyz10, const float* __restrict__ w1,
                                      const float* __restrict__ g1, const float* __restrict__ b1,
                                      const float* __restrict__ w2, const float* __restrict__ b2,
                                      const float* __restrict__ stats, float* __restrict__ sc_out,
                                      int total) {
  int r = blockIdx.x * blockDim.x + threadIdx.x;
  if (r >= total) return;
  const float inv = 1.f / (float)(ROWS * Kn);
  const float* xr = xyz10 + (size_t)r * 10;
  float hh[16];
  for (int o = 0; o < 16; ++o) {
    float h = 0.f;
    for (int c = 0; c < 10; ++c) h += xr[c] * w1[o*10 + c];
    float mu = stats[o] * inv;
    float var = stats[16+o] * inv - mu*mu;
    float s_ = g1[o] * rsqrtf(var + 1e-5f);
    h = (h - mu) * s_ + b1[o];
    hh[o] = h > 0.f ? h : 0.f;
  }
  float s[8]; float mx = NEG_INF;
  for (int m = 0; m < 8; ++m) {
    float v = b2[m];
    for (int o = 0; o < 16; ++o) v += hh[o] * w2[m*16 + o];
    s[m] = v; mx = v > mx ? v : mx;
  }
  float sum = 0.f;
  for (int m = 0; m < 8; ++m) { s[m] = expf(s[m] - mx); sum += s[m]; }
  float is = 1.f / sum;
  float* out = sc_out + (size_t)r * 8;
  for (int m = 0; m < 8; ++m) out[m] = s[m] * is;
}

extern "C" __global__ void k_scsum(const float* __restrict__ sc, float* __restrict__ scsum) {
  int row = blockIdx.x * blockDim.x + threadIdx.x;
  if (row >= ROWS) return;
  for (int m = 0; m < 8; ++m) {
    float a = 0.f;
    for (int k = 0; k < Kn; ++k) a += sc[((size_t)row * Kn + k) * 8 + m];
    scsum[(size_t)row * 8 + m] = a;
  }
}

// Build transposed weight banks: wt[n][k] = mat[k][n] + mat[k+64][n],  ct[n][k] = mat[k][n]
// (cat(f,f)@mat == f@Wsum; transposed to (out,in) so GEMM B-loads are contiguous float2)
extern "C" __global__ void k_wprep(const float* __restrict__ mat, float* __restrict__ wt,
                                   float* __restrict__ ct) {
  int g = blockIdx.x * blockDim.x + threadIdx.x;
  if (g >= 64 * 512) return;
  int n = g >> 6, k = g & 63;
  float top = mat[k * 512 + n];
  ct[g] = top;
  wt[g] = top + mat[(k + 64) * 512 + n];
}

// BN scale/shift precompute for fused GEMM A-loads: ss[c]=scale, ss[cols+c]=shift
extern "C" __global__ void k_bnprep(const float* __restrict__ stats, int cols, float invc,
                                    const float* __restrict__ g, const float* __restrict__ be,
                                    float* __restrict__ ss) {
  int c = blockIdx.x * blockDim.x + threadIdx.x;
  if (c >= cols) return;
  float mu = stats[c] * invc;
  float var = stats[cols + c] * invc - mu*mu;
  float s_ = g[c] * rsqrtf(var + 1e-5f);
  ss[c] = s_;
  ss[cols + c] = be[c] - mu * s_;
}

// ---------------- generic f32 WMMA GEMM:  C[M,N] = A[M,K] * B[N,K]^T ----------------
// Each wave computes a 16x32 tile (two 16x16 WMMA accumulators sharing the A fragment).
// AMODE: 0 = plain A; 1 = relu(A*scale+shift) from precomputed ss; 2 = concat(xx[320], clsfeat[1088])
template <int AMODE>
__global__ __launch_bounds__(256) void k_gemm(const float* __restrict__ A, int lda,
                    const float* __restrict__ Bw, int ldb,
                    float* __restrict__ C, int ldc,
                    int rowsM, int colsN, int Kd,
                    const float* __restrict__ ss,
                    const float* __restrict__ clsfeat,
                    const float* __restrict__ bias) {
  int wave = (blockIdx.x * blockDim.x + threadIdx.x) >> 5;
  int lane = threadIdx.x & 31;
  int tilesN = (colsN + 15) >> 4;
  int pairs  = (tilesN + 1) >> 1;
  int tilesM = rowsM >> 4;
  if (wave >= tilesM * pairs) return;        // wave-uniform exit
  int tm = wave / pairs, tp = wave % pairs;
  int r = lane & 15, half = lane >> 4;
  int am  = tm * 16 + r;                     // A row for this lane
  int bn0 = tp * 32 + r;                     // B col, tile 0
  int bn1 = bn0 + 16;                        // B col, tile 1
  float zm0 = bn0 < colsN ? 1.f : 0.f;
  float zm1 = bn1 < colsN ? 1.f : 0.f;
  int bn0c = bn0 < colsN ? bn0 : 0;
  int bn1c = bn1 < colsN ? bn1 : 0;
  const float* arow  = A + (size_t)am * lda;
  const float* crow  = (AMODE == 2) ? clsfeat + (size_t)(am >> 11) * 1088 : nullptr;
  const float* brow0 = Bw + (size_t)bn0c * ldb;
  const float* brow1 = Bw + (size_t)bn1c * ldb;
  v8f acc0 = {0.f,0.f,0.f,0.f,0.f,0.f,0.f,0.f};
  v8f acc1 = {0.f,0.f,0.f,0.f,0.f,0.f,0.f,0.f};
  int kbase = 2 * half;                      // frag element v holds K = k0 + v + 2*half
#pragma unroll 4
  for (int k0 = 0; k0 < Kd; k0 += 4) {
    int ka = k0 + kbase;
    v2f a;
    if (AMODE == 2) {
      a = (ka < 320) ? *(const v2f*)(arow + ka) : *(const v2f*)(crow + (ka - 320));
    } else {
      a = *(const v2f*)(arow + ka);
      if (AMODE == 1) {
        v2f scv = *(const v2f*)(ss + ka);
        v2f shv = *(const v2f*)(ss + Kd + ka);
        a = a * scv + shv;
        a[0] = a[0] > 0.f ? a[0] : 0.f;
        a[1] = a[1] > 0.f ? a[1] : 0.f;
      }
    }
    v2f b0 = *(const v2f*)(brow0 + ka) * zm0;
    v2f b1 = *(const v2f*)(brow1 + ka) * zm1;
    acc0 = __builtin_amdgcn_wmma_f32_16x16x4_f32(false, a, false, b0, (short)0, acc0, false, false);
    acc1 = __builtin_amdgcn_wmma_f32_16x16x4_f32(false, a, false, b1, (short)0, acc1, false, false);
  }
  int cn0 = tp * 32 + (lane & 15);
  int cn1 = cn0 + 16;
  if (cn0 < colsN) {
    float bb = bias ? bias[cn0] : 0.f;
    for (int v = 0; v < 8; ++v) {
      int cm = tm * 16 + v + 8 * half;       // C/D layout: VGPR v + lane-half -> M
      C[(size_t)cm * ldc + cn0] = acc0[v] + bb;
    }
  }
  if (cn1 < colsN) {
    float bb = bias ? bias[cn1] : 0.f;
    for (int v = 0; v < 8; ++v) {
      int cm = tm * 16 + v + 8 * half;
      C[(size_t)cm * ldc + cn1] = acc1[v] + bb;
    }
  }
}

// ---------------- paconv assembly: one wave per (b,n); lane owns o=2l,2l+1 ----------------
extern "C" __global__ void k_assemble(const float* __restrict__ sc, const float* __restrict__ scsum,
                                      const int* __restrict__ idx, const float* __restrict__ po,
                                      const float* __restrict__ co, float* __restrict__ pre) {
  int wave = (blockIdx.x * blockDim.x + threadIdx.x) >> 5;
  int lane = threadIdx.x & 31;
  if (wave >= ROWS) return;
  int b = wave >> 11;
  int o0 = lane * 2;
  float a0 = 0.f, a1 = 0.f;
  for (int k = 0; k < Kn; ++k) {
    int j = idx[(size_t)wave * Kn + k];
    const float* srow = sc + ((size_t)wave * Kn + k) * 8;
    const float* prow = po + ((size_t)(b * Npts + j)) * 512;
    for (int m = 0; m < 8; ++m) {
      float s = srow[m];
      v2f p = *(const v2f*)(prow + m*64 + o0);
      a0 += s * p[0];
      a1 += s * p[1];
    }
  }
  const float* crow = co + (size_t)wave * 512;
  const float* ssr  = scsum + (size_t)wave * 8;
  for (int m = 0; m < 8; ++m) {
    float s = ssr[m];
    v2f c = *(const v2f*)(crow + m*64 + o0);
    a0 -= s * c[0];
    a1 -= s * c[1];
  }
  pre[(size_t)wave * 64 + o0]     = a0;
  pre[(size_t)wave * 64 + o0 + 1] = a1;
}

// ---------------- generic per-column stats / BN apply ----------------
extern "C" __global__ void k_colstats(const float* __restrict__ X, int rows, int cols, int ldx,
                                      float* __restrict__ stats) {
  __shared__ float ssum[256], ssq[256];
  int c = blockIdx.x, t = threadIdx.x;
  float sum = 0.f, sq = 0.f;
  for (int r = t; r < rows; r += 256) {
    float v = X[(size_t)r * ldx + c];
    sum += v; sq += v*v;
  }
  ssum[t] = sum; ssq[t] = sq;
  __syncthreads();
  for (int s = 128; s > 0; s >>= 1) {
    if (t < s) { ssum[t] += ssum[t+s]; ssq[t] += ssq[t+s]; }
    __syncthreads();
  }
  if (t == 0) { stats[c] = ssum[0]; stats[cols + c] = ssq[0]; }
}

extern "C" __global__ void k_applybn(const float* __restrict__ X, int rows, int cols, int ldx,
                                     const float* __restrict__ stats, float invc,
                                     const float* __restrict__ g, const float* __restrict__ be,
                                     float* __restrict__ Y, int ldy, int offy, int total) {
  int gi = blockIdx.x * blockDim.x + threadIdx.x;
  if (gi >= total) return;
  int r = gi / cols, c = gi % cols;
  float mu = stats[c] * invc;
  float var = stats[cols + c] * invc - mu*mu;
  float s_ = g[c] * rsqrtf(var + 1e-5f);
  float v = (X[(size_t)r * ldx + c] - mu) * s_ + be[c];
  Y[(size_t)r * ldy + offy + c] = v > 0.f ? v : 0.f;
}

// ---------------- global feature max-pool + class head ----------------
extern "C" __global__ void k_xtmax(const float* __restrict__ ht, const float* __restrict__ stats,
                                   const float* __restrict__ g, const float* __restrict__ be,
                                   float* __restrict__ clsfeat) {
  __shared__ float sm[256];
  int blk = blockIdx.x;
  int b = blk >> 10, c = blk & 1023, t = threadIdx.x;
  const float invc = 1.f / (float)ROWS;
  float mu = stats[c] * invc;
  float var = stats[1024 + c] * invc - mu*mu;
  float s_ = g[c] * rsqrtf(var + 1e-5f);
  float sh = be[c] - mu * s_;
  float mx = NEG_INF;
  for (int n = t; n < Npts; n += 256) {
    float v = ht[((size_t)(b * Npts + n)) * 1024 + c] * s_ + sh;
    v = v > 0.f ? v : 0.f;
    mx = v > mx ? v : mx;
  }
  sm[t] = mx;
  __syncthreads();
  for (int s = 128; s > 0; s >>= 1) {
    if (t < s) sm[t] = sm[t] > sm[t+s] ? sm[t] : sm[t+s];
    __syncthreads();
  }
  if (t == 0) clsfeat[(size_t)b * 1088 + c] = sm[0];
}

extern "C" __global__ void k_clshead(const float* __restrict__ cls, const float* __restrict__ wc,
                                     const float* __restrict__ g, const float* __restrict__ be,
                                     float* __restrict__ clsfeat) {
  int o = threadIdx.x;
  if (o >= 64) return;
  float v[Bsz]; float mu = 0.f;
  for (int b = 0; b < Bsz; ++b) {
    float a = 0.f;
    for (int c = 0; c < 16; ++c) a += cls[b*16 + c] * wc[o*16 + c];
    v[b] = a; mu += a;
  }
  mu *= (1.f / Bsz);
  float var = 0.f;
  for (int b = 0; b < Bsz; ++b) { float d = v[b] - mu; var += d*d; }
  var *= (1.f / Bsz);
  float s_ = g[o] * rsqrtf(var + 1e-5f);
  for (int b = 0; b < Bsz; ++b) {
    float y = (v[b] - mu) * s_ + be[o];
    clsfeat[(size_t)b * 1088 + 1024 + o] = y > 0.f ? y : 0.f;
  }
}

extern "C" __global__ void k_logsoftmax(float* __restrict__ out, int rows, int cols) {
  int r = blockIdx.x * blockDim.x + threadIdx.x;
  if (r >= rows) return;
  float* x = out + (size_t)r * cols;
  float mx = NEG_INF;
  for (int c = 0; c < cols; ++c) mx = x[c] > mx ? x[c] : mx;
  float s = 0.f;
  for (int c = 0; c < cols; ++c) s += expf(x[c] - mx);
  float lse = mx + logf(s);
  for (int c = 0; c < cols; ++c) x[c] -= lse;
}

// ---------------- host-side GEMM launch helper ----------------
template <int AMODE>
static inline void launch_gemm(const float* A, int lda, const float* Bw, int ldb,
                               float* C, int ldc, int rowsM, int colsN, int Kd,
                               const float* ss, const float* clsfeat, const float* bias,
                               hipStream_t stream) {
  int tilesN = (colsN + 15) / 16;
  int pairs  = (tilesN + 1) / 2;
  int tiles  = (rowsM / 16) * pairs;
  int blocks = (tiles * 32 + 255) / 256;
  k_gemm<AMODE><<<blocks, 256, 0, stream>>>(A, lda, Bw, ldb, C, ldc, rowsM, colsN, Kd,
                                            ss, clsfeat, bias);
}

extern "C" void kernel_launch(void* const* d_in, const int* in_sizes, int n_in,
                              void* d_out, int out_size, void* d_ws, size_t ws_size,
                              hipStream_t stream) {
  float* ws = (float*)d_ws;
  const float* x = (const float*)d_in[0];
  const float* cls_label = (const float*)d_in[2];
  auto P = [&](int i) { return (const float*)d_in[i]; };
  // param indices (dict order): 3 conv1_w, 4 conv1_b, 5 bn1_g, 6 bn1_b,
  // per layer i=2..5 (base 7+(i-2)*8): w1,g1,b1,w2,b2,mat,bn_g,bn_b,
  // 39 convt_w, 40 bnt_g, 41 bnt_b, 42 convc_w, 43 bnc_g, 44 bnc_b,
  // 45 conv6_w, 46 bn6_g, 47 bn6_b, 48 conv7_w, 49 bn7_g, 50 bn7_b,
  // 51 conv8_w, 52 bn8_g, 53 bn8_b, 54 conv9_w, 55 conv9_b

  float* pts     = ws + OFF_PTS;
  int*   idx     = (int*)(ws + OFF_IDX);
  float* xyz10   = ws + OFF_XYZ10;
  float* sc      = ws + OFF_SC;
  float* scsum   = ws + OFF_SCSUM;
  float* po      = ws + OFF_PO;
  float* co      = ws + OFF_CO;
  float* pre     = ws + OFF_PRE;
  float* xx      = ws + OFF_XX;
  float* ht      = ws + OFF_HT;
  float* clsfeat = ws + OFF_CLS;
  float* h6      = ws + OFF_H6;
  float* h7      = ws + OFF_H7;
  float* h8      = ws + OFF_H8;
  float* stats   = ws + OFF_STATS;
  float* ssbuf   = ws + OFF_SS;
  float* wt      = ws + OFF_WT;
  float* ct      = ws + OFF_CT;

  const float invBN = 1.f / (float)ROWS;

  // graph construction
  k_transpose<<<96, 256, 0, stream>>>(x, pts, Bsz * 3 * Npts);
  k_knn<<<ROWS, 256, 0, stream>>>(pts, idx);
  k_xyz10<<<960, 256, 0, stream>>>(pts, idx, xyz10, ROWS * Kn);

  // conv1 edge-conv -> x1 (xx[:, 0:64])
  k_conv1_stats<<<64, 256, 0, stream>>>(xyz10, P(3), P(4), stats + ST_BN1);
  k_conv1_apply<<<32, 256, 0, stream>>>(xyz10, P(3), P(4), stats + ST_BN1, P(5), P(6), xx);

  // paconv layers 2..5
  for (int i = 2; i <= 5; ++i) {
    int pb = 7 + (i - 2) * 8;
    const float *snw1 = P(pb), *sng1 = P(pb+1), *snb1 = P(pb+2), *snw2 = P(pb+3), *snb2 = P(pb+4);
    const float *mat = P(pb+5), *bng = P(pb+6), *bnb = P(pb+7);
    k_sn_stats<<<16, 256, 0, stream>>>(xyz10, snw1, stats + ST_SN);
    k_sn_apply<<<960, 256, 0, stream>>>(xyz10, snw1, sng1, snb1, snw2, snb2, stats + ST_SN, sc, ROWS * Kn);
    k_scsum<<<32, 256, 0, stream>>>(sc, scsum);
    k_wprep<<<128, 256, 0, stream>>>(mat, wt, ct);
    // po = feat @ Wsum^T   (feat = xx channel slice, Kd=64; B now (out=512,in=64) row-major)
    launch_gemm<0>(xx + 64 * (i - 2), 320, wt, 64, po, 512, ROWS, 512, 64,
                   nullptr, nullptr, nullptr, stream);
    // co = feat @ kern_top^T
    launch_gemm<0>(xx + 64 * (i - 2), 320, ct, 64, co, 512, ROWS, 512, 64,
                   nullptr, nullptr, nullptr, stream);
    k_assemble<<<1024, 256, 0, stream>>>(sc, scsum, idx, po, co, pre);
    k_colstats<<<64, 256, 0, stream>>>(pre, ROWS, 64, 64, stats + ST_PA);
    k_applybn<<<2048, 256, 0, stream>>>(pre, ROWS, 64, 64, stats + ST_PA, invBN, bng, bnb,
                                        xx, 320, 64 * (i - 1), ROWS * 64);
  }

  // convt: (8192,320)@(320,1024); weight already (out,in)
  launch_gemm<0>(xx, 320, P(39), 320, ht, 1024, ROWS, 1024, 320,
                 nullptr, nullptr, nullptr, stream);
  k_colstats<<<1024, 256, 0, stream>>>(ht, ROWS, 1024, 1024, stats + ST_T);
  k_xtmax<<<Bsz * 1024, 256, 0, stream>>>(ht, stats + ST_T, P(40), P(41), clsfeat);
  k_clshead<<<1, 64, 0, stream>>>(cls_label, P(42), P(43), P(44), clsfeat);

  // conv6: A = concat(xx, broadcast clsfeat) (AMODE 2), Kd=1408
  launch_gemm<2>(xx, 320, P(45), 1408, h6, 256, ROWS, 256, 1408,
                 nullptr, clsfeat, nullptr, stream);
  k_colstats<<<256, 256, 0, stream>>>(h6, ROWS, 256, 256, stats + ST_6);
  // conv7: A = relu(bn6(h6)) fused via precomputed scale/shift
  k_bnprep<<<1, 256, 0, stream>>>(stats + ST_6, 256, invBN, P(46), P(47), ssbuf);
  launch_gemm<1>(h6, 256, P(48), 256, h7, 256, ROWS, 256, 256,
                 ssbuf, nullptr, nullptr, stream);
  k_colstats<<<256, 256, 0, stream>>>(h7, ROWS, 256, 256, stats + ST_7);
  // conv8
  k_bnprep<<<1, 256, 0, stream>>>(stats + ST_7, 256, invBN, P(49), P(50), ssbuf);
  launch_gemm<1>(h7, 256, P(51), 256, h8, 128, ROWS, 128, 256,
                 ssbuf, nullptr, nullptr, stream);
  k_colstats<<<128, 256, 0, stream>>>(h8, ROWS, 128, 128, stats + ST_8);
  // conv9 (+bias) straight into d_out
  k_bnprep<<<1, 128, 0, stream>>>(stats + ST_8, 128, invBN, P(52), P(53), ssbuf);
  launch_gemm<1>(h8, 128, P(54), 128, (float*)d_out, 50, ROWS, 50, 128,
                 ssbuf, nullptr, P(55), stream);
  k_logsoftmax<<<32, 256, 0, stream>>>((float*)d_out, ROWS, 50);
}